// GRCGNN_49366354100286
// MI455X (gfx1250) — compile-verified
//
#include <hip/hip_runtime.h>

// Problem constants (match reference).
#define NN   50000
#define EE   1600000
#define INF  385
#define HH   128

typedef __attribute__((ext_vector_type(2))) float v2f;
typedef __attribute__((ext_vector_type(8))) float v8f;

// ---------------------------------------------------------------- utilities
__global__ void zero_f32(float* __restrict__ p, long long n) {
    long long i = (long long)blockIdx.x * blockDim.x + threadIdx.x;
    if (i < n) p[i] = 0.0f;
}

__global__ void relu_inplace(float* __restrict__ p, long long n) {
    long long i = (long long)blockIdx.x * blockDim.x + threadIdx.x;
    if (i < n) { float v = p[i]; p[i] = v > 0.0f ? v : 0.0f; }
}

__global__ void relu_copy(float* __restrict__ d, const float* __restrict__ s, long long n) {
    long long i = (long long)blockIdx.x * blockDim.x + threadIdx.x;
    if (i < n) { float v = s[i]; d[i] = v > 0.0f ? v : 0.0f; }
}

// -------------------------------------------------------- degree computation
// Counts accumulated as u32 in the rs buffers, then transformed in place to
// rsqrt(max(deg,1)). Recomputed every call (stateless / capture-safe).
__global__ void count_deg(const int* __restrict__ src, const int* __restrict__ dst,
                          unsigned* __restrict__ cnt_out, unsigned* __restrict__ cnt_in) {
    long long i = (long long)blockIdx.x * blockDim.x + threadIdx.x;
    if (i >= (long long)5 * EE) return;
    int k = (int)(i / EE);
    atomicAdd(cnt_out + (size_t)k * NN + src[i], 1u);
    atomicAdd(cnt_in  + (size_t)k * NN + dst[i], 1u);
}

__global__ void deg_to_rsqrt(void* __restrict__ buf, long long n) {
    long long i = (long long)blockIdx.x * blockDim.x + threadIdx.x;
    if (i >= n) return;
    unsigned c = ((const unsigned*)buf)[i];
    float d = (float)c;
    ((float*)buf)[i] = rsqrtf(d < 1.0f ? 1.0f : d);
}

// ------------------------------------------------------------- edge scatter
// One wave per edge: 32 lanes each move a float4 chunk of the 128-wide row.
// scale = rsqrt(deg_out[s]) * rsqrt(deg_in[d]) folded per edge, so
//   agg[d] += h[s] * scale   ==   (D_in^-1/2 A D_out^-1/2 X)[d]
__global__ void scatter_edges(const float* __restrict__ h,
                              const int* __restrict__ src,
                              const int* __restrict__ dst,
                              const float* __restrict__ rs_out,
                              const float* __restrict__ rs_in,
                              float* __restrict__ agg) {
    long long idx = (long long)blockIdx.x * blockDim.x + threadIdx.x;
    if (idx >= (long long)EE * 32) return;
    int e = (int)(idx >> 5);
    int c = (int)(idx & 31);          // wave-uniform e, per-lane chunk c
    int s = src[e];
    int d = dst[e];
    float scale = rs_out[s] * rs_in[d];
    const float4 v = *(const float4*)(h + (size_t)s * HH + c * 4);
    float* o = agg + (size_t)d * HH + c * 4;
    unsafeAtomicAdd(o + 0, v.x * scale);
    unsafeAtomicAdd(o + 1, v.y * scale);
    unsafeAtomicAdd(o + 2, v.z * scale);
    unsafeAtomicAdd(o + 3, v.w * scale);
}

// ------------------------------------------------------------- WMMA GEMM
// C[M x 128] = op(A[M x K(lda)] @ B[K x 128] + bias), fp32 via
// V_WMMA_F32_16X16X4_F32. One wave owns a 16-row strip and all 8 column
// tiles: A fragment loaded once per K-step, reused by 8 WMMAs.
// Fragment layouts per ISA 7.12.2:
//   A 16x4:  lane<16 -> M=lane, {K0,K1}; lane>=16 -> M=lane-16, {K2,K3}
//   B 4x16:  lane<16 -> N=lane, {K0,K1}; lane>=16 -> N=lane-16, {K2,K3}
//   C 16x16: vgpr j, lane L -> row = j + 8*(L>>4), col = L&15
__global__ void __launch_bounds__(128)
gemm_f32_wmma(const float* __restrict__ A, int lda,
              const float* __restrict__ B,      // [K][128] row-major
              const float* __restrict__ bias,   // [128]
              float* __restrict__ C,            // [M][128]
              int M, int K, int relu, int accum) {
    const int lane = threadIdx.x & 31;
    const int wave = blockIdx.x * (blockDim.x >> 5) + (threadIdx.x >> 5);
    const int nstrips = M >> 4;                 // M % 16 == 0 here
    if (wave >= nstrips) return;                // wave-uniform: EXEC stays all-1
    const int m0  = wave << 4;
    const int ln  = lane & 15;
    const int hi  = lane >> 4;                  // 0: K0/K1 half, 1: K2/K3 half
    const int row = m0 + ln;

    v8f acc[8] = {};
    const float* Arow = A + (size_t)row * lda;
    const int kmain = K & ~3;

    for (int k0 = 0; k0 < kmain; k0 += 4) {
        const int kk = k0 + (hi << 1);
        v2f a; a.x = Arow[kk]; a.y = Arow[kk + 1];
        const float* B0 = B + (size_t)kk * HH;
#pragma unroll
        for (int t = 0; t < 8; ++t) {
            const int col = t * 16 + ln;
            v2f b; b.x = B0[col]; b.y = B0[HH + col];
            acc[t] = __builtin_amdgcn_wmma_f32_16x16x4_f32(
                false, a, false, b, (short)0, acc[t], false, false);
        }
    }
    if (kmain < K) {                            // K=385 tail (zero-padded)
        const int kk = kmain + (hi << 1);
        v2f a;
        a.x = (kk     < K) ? Arow[kk]     : 0.0f;
        a.y = (kk + 1 < K) ? Arow[kk + 1] : 0.0f;
#pragma unroll
        for (int t = 0; t < 8; ++t) {
            const int col = t * 16 + ln;
            v2f b;
            b.x = (kk     < K) ? B[(size_t)kk * HH + col]       : 0.0f;
            b.y = (kk + 1 < K) ? B[(size_t)(kk + 1) * HH + col] : 0.0f;
            acc[t] = __builtin_amdgcn_wmma_f32_16x16x4_f32(
                false, a, false, b, (short)0, acc[t], false, false);
        }
    }

#pragma unroll
    for (int t = 0; t < 8; ++t) {
        const int col = t * 16 + ln;
        const float bv = bias[col];
#pragma unroll
        for (int j = 0; j < 8; ++j) {
            const size_t idx = (size_t)(m0 + hi * 8 + j) * HH + col;
            float v = acc[t][j] + bv;
            if (accum) v += C[idx];
            if (relu)  v = v > 0.0f ? v : 0.0f;
            C[idx] = v;
        }
    }
}

// ---------------------------------------------------------------- launcher
extern "C" void kernel_launch(void* const* d_in, const int* in_sizes, int n_in,
                              void* d_out, int out_size, void* d_ws, size_t ws_size,
                              hipStream_t stream) {
    (void)in_sizes; (void)n_in; (void)out_size; (void)ws_size;
    const float* feat[4] = {(const float*)d_in[0], (const float*)d_in[1],
                            (const float*)d_in[2], (const float*)d_in[3]};
    const int*   src = (const int*)d_in[4];     // [5][E]
    const int*   dst = (const int*)d_in[5];     // [5][E]
    const float* Wp  = (const float*)d_in[6];   // [4][385][128]
    const float* bp  = (const float*)d_in[7];   // [4][128]
    const float* Wc1 = (const float*)d_in[8];   // [5][128][128]
    const float* bc1 = (const float*)d_in[9];   // [5][128]
    const float* Wc2 = (const float*)d_in[10];
    const float* bc2 = (const float*)d_in[11];
    float* out = (float*)d_out;                 // [4][N][128]

    const size_t NH = (size_t)NN * HH;
    float* ws     = (float*)d_ws;
    float* hbuf   = ws;                         // 4*NH : h0, later layer-2 out
    float* agg    = ws + 4 * NH;                // NH   : per-etype aggregate
    float* rs_out = ws + 5 * NH;                // 5*NN : rsqrt(out-degree)
    float* rs_in  = rs_out + 5 * (size_t)NN;    // 5*NN : rsqrt(in-degree)
    // ws footprint: (5*NH + 10*NN)*4 B = 130.0 MB

    const int ST[5] = {0, 1, 3, 0, 2};          // ETYPES src types
    const int DT[5] = {1, 3, 2, 2, 0};          // ETYPES dst types

    // ---- normalization factors (recomputed every call)
    { long long n = 10LL * NN;
      zero_f32<<<(unsigned)((n + 255) / 256), 256, 0, stream>>>(rs_out, n); }
    { long long n = 5LL * EE;
      count_deg<<<(unsigned)((n + 255) / 256), 256, 0, stream>>>(
          src, dst, (unsigned*)rs_out, (unsigned*)rs_in); }
    { long long n = 10LL * NN;
      deg_to_rsqrt<<<(unsigned)((n + 255) / 256), 256, 0, stream>>>(rs_out, n); }

    const unsigned gemm_blocks = (NN / 16 + 3) / 4;   // 4 waves/block, 3125 strips
    const unsigned scat_blocks = (unsigned)(((long long)EE * 32 + 255) / 256);
    const unsigned zero_blocks = (unsigned)((NH + 255) / 256);

    // ---- input projection: h0[i] = relu(feat[i] @ Wp[i] + bp[i])
    for (int i = 0; i < 4; ++i)
        gemm_f32_wmma<<<gemm_blocks, 128, 0, stream>>>(
            feat[i], INF, Wp + (size_t)i * INF * HH, bp + (size_t)i * HH,
            hbuf + (size_t)i * NH, NN, INF, /*relu=*/1, /*accum=*/0);

    // ---- layer 1: d_out[dt] (+)= (scatter(h0[st]) @ Wc1[k] + bc1[k]); relu
    for (int k = 0; k < 5; ++k) {
        zero_f32<<<zero_blocks, 256, 0, stream>>>(agg, (long long)NH);
        scatter_edges<<<scat_blocks, 256, 0, stream>>>(
            hbuf + (size_t)ST[k] * NH, src + (size_t)k * EE, dst + (size_t)k * EE,
            rs_out + (size_t)k * NN, rs_in + (size_t)k * NN, agg);
        gemm_f32_wmma<<<gemm_blocks, 128, 0, stream>>>(
            agg, HH, Wc1 + (size_t)k * HH * HH, bc1 + (size_t)k * HH,
            out + (size_t)DT[k] * NH, NN, HH, /*relu=*/0, /*accum=*/(k == 3) ? 1 : 0);
    }
    relu_inplace<<<(unsigned)((4 * NH + 255) / 256), 256, 0, stream>>>(out, 4LL * NH);

    // ---- layer 2: hbuf[dt] (+)= (scatter(h1[st]) @ Wc2[k] + bc2[k])
    for (int k = 0; k < 5; ++k) {
        zero_f32<<<zero_blocks, 256, 0, stream>>>(agg, (long long)NH);
        scatter_edges<<<scat_blocks, 256, 0, stream>>>(
            out + (size_t)ST[k] * NH, src + (size_t)k * EE, dst + (size_t)k * EE,
            rs_out + (size_t)k * NN, rs_in + (size_t)k * NN, agg);
        gemm_f32_wmma<<<gemm_blocks, 128, 0, stream>>>(
            agg, HH, Wc2 + (size_t)k * HH * HH, bc2 + (size_t)k * HH,
            hbuf + (size_t)DT[k] * NH, NN, HH, /*relu=*/0, /*accum=*/(k == 3) ? 1 : 0);
    }

    // ---- final relu into d_out
    relu_copy<<<(unsigned)((4 * NH + 255) / 256), 256, 0, stream>>>(out, hbuf, 4LL * NH);
}